// ScaleDotProductAttention_28398323761673
// MI455X (gfx1250) — compile-verified
//
#include <hip/hip_runtime.h>

typedef float v2f __attribute__((ext_vector_type(2)));
typedef float v8f __attribute__((ext_vector_type(8)));

#define NEG_E (-1e12f)

// Reduce across the 16 lanes of each wave32 half (xor masks < 16 stay in-half).
__device__ __forceinline__ float rowmax16(float v) {
#pragma unroll
  for (int off = 1; off < 16; off <<= 1)
    v = fmaxf(v, __shfl_xor(v, off, 32));
  return v;
}
__device__ __forceinline__ float rowsum16(float v) {
#pragma unroll
  for (int off = 1; off < 16; off <<= 1)
    v += __shfl_xor(v, off, 32);
  return v;
}

// bs=4, h=16, s=2048, d=64, fp32.  One wave handles a 16-row query tile.
// 8 waves per workgroup -> 128 query rows per block.
__global__ __launch_bounds__(256)
void sdpa_f32_wmma_kernel(const float* __restrict__ q,
                          const float* __restrict__ k,
                          const float* __restrict__ v,
                          const int*   __restrict__ mask,
                          float* __restrict__ out,
                          float* __restrict__ score) {
  constexpr int S = 2048, D = 64, H = 16;
  constexpr float SCALE = 0.125f;  // 1/sqrt(64)

  // Per-wave P-tile staging for the C-layout -> A-layout transpose.
  // Stride 17 avoids LDS bank conflicts between the two wave halves.
  __shared__ float lds_p[8][16 * 17];

  const int lane = threadIdx.x & 31;
  const int wave = threadIdx.x >> 5;
  const int m  = lane & 15;  // A-frag row / B-frag column / C-tile column
  const int h2 = lane >> 4;  // wave-half selector
  const int b  = blockIdx.z;
  const int hh = blockIdx.y;
  const int q0 = (blockIdx.x * 8 + wave) * 16;

  const size_t bh = (size_t)b * H + hh;
  const float* Q = q + bh * S * D;
  const float* K = k + bh * S * D;
  const float* V = v + bh * S * D;
  const int*   M = mask + (size_t)b * S * S;
  float* sc = score + bh * S * S;
  float* P  = &lds_p[wave][0];

  // ---- Preload Q tile as 16 WMMA-A fragments (16x4 f32 each) ----
  // A layout: lane m = row, VGPR0 = K=2*h2, VGPR1 = K=2*h2+1 -> contiguous float2.
  v2f aq[16];
#pragma unroll
  for (int f = 0; f < 16; ++f)
    aq[f] = *(const v2f*)&Q[(size_t)(q0 + m) * D + f * 4 + 2 * h2];

  // Running softmax stats: lane holds rows r + 8*h2 (replicated across its half).
  float mrow[8], srow[8];
#pragma unroll
  for (int r = 0; r < 8; ++r) { mrow[r] = -INFINITY; srow[r] = 0.0f; }

  // ================= Pass 1: row max / row sum =================
  for (int kt = 0; kt < S / 16; ++kt) {
    const int k0 = kt * 16;
    if (kt + 1 < S / 16)
      __builtin_prefetch(&K[(size_t)(k0 + 16 + m) * D + 2 * h2], 0, 0);

    v8f acc = {};
#pragma unroll
    for (int f = 0; f < 16; ++f) {
      // B = K^T fragment: lane n = k-row, contiguous float2 along d.
      v2f bk = *(const v2f*)&K[(size_t)(k0 + m) * D + f * 4 + 2 * h2];
      acc = __builtin_amdgcn_wmma_f32_16x16x4_f32(false, aq[f], false, bk,
                                                  (short)0, acc, false, false);
    }
#pragma unroll
    for (int r = 0; r < 8; ++r) {
      const int row = r + 8 * h2;
      const int mk  = M[(size_t)(q0 + row) * S + (k0 + m)];
      const float val  = (mk == 0) ? NEG_E : acc[r] * SCALE;
      const float mnew = fmaxf(mrow[r], rowmax16(val));
      const float e    = __expf(val - mnew);
      srow[r] = srow[r] * __expf(mrow[r] - mnew) + rowsum16(e);
      mrow[r] = mnew;
    }
  }
  float inv[8];
#pragma unroll
  for (int r = 0; r < 8; ++r) inv[r] = 1.0f / srow[r];

  // ================= Pass 2: normalize, emit score, O = P*V =================
  v8f oacc[4] = {};  // O tile 16x64 as 4 accumulators of 16x16
  for (int kt = 0; kt < S / 16; ++kt) {
    const int k0 = kt * 16;
    if (kt + 1 < S / 16) {
      __builtin_prefetch(&K[(size_t)(k0 + 16 + m) * D + 2 * h2], 0, 0);
      __builtin_prefetch(&V[(size_t)(k0 + 16 + m) * D + 2 * h2], 0, 0);
    }

    v8f acc = {};
#pragma unroll
    for (int f = 0; f < 16; ++f) {
      v2f bk = *(const v2f*)&K[(size_t)(k0 + m) * D + f * 4 + 2 * h2];
      acc = __builtin_amdgcn_wmma_f32_16x16x4_f32(false, aq[f], false, bk,
                                                  (short)0, acc, false, false);
    }

    // Normalize; write softmax output (streaming -> non-temporal) and stage in LDS.
#pragma unroll
    for (int r = 0; r < 8; ++r) {
      const int row = r + 8 * h2;
      const int mk  = M[(size_t)(q0 + row) * S + (k0 + m)];
      const float val = (mk == 0) ? NEG_E : acc[r] * SCALE;
      const float p   = __expf(val - mrow[r]) * inv[r];
      __builtin_nontemporal_store(p, &sc[(size_t)(q0 + row) * S + (k0 + m)]);
      P[row * 17 + m] = p;
    }
    // DS ops are in-order within the wave; wait so cross-lane reads see the stores.
    asm volatile("s_wait_dscnt 0" ::: "memory");

    // O += P(16x16) x V(16x64): 4 K-chunks x 4 N-tiles of v_wmma_f32_16x16x4_f32.
#pragma unroll
    for (int c = 0; c < 4; ++c) {
      v2f ap;
      ap.x = P[m * 17 + 4 * c + 2 * h2];
      ap.y = P[m * 17 + 4 * c + 2 * h2 + 1];
#pragma unroll
      for (int nt = 0; nt < 4; ++nt) {
        v2f bv;
        bv.x = V[(size_t)(k0 + 4 * c + 2 * h2) * D + nt * 16 + m];
        bv.y = V[(size_t)(k0 + 4 * c + 2 * h2 + 1) * D + nt * 16 + m];
        oacc[nt] = __builtin_amdgcn_wmma_f32_16x16x4_f32(false, ap, false, bv,
                                                         (short)0, oacc[nt],
                                                         false, false);
      }
    }
  }

  // ---- Write O tile ----
#pragma unroll
  for (int nt = 0; nt < 4; ++nt)
#pragma unroll
    for (int r = 0; r < 8; ++r)
      __builtin_nontemporal_store(
          oacc[nt][r],
          &out[(bh * S + q0 + r + 8 * h2) * D + nt * 16 + m]);
}

extern "C" void kernel_launch(void* const* d_in, const int* in_sizes, int n_in,
                              void* d_out, int out_size, void* d_ws, size_t ws_size,
                              hipStream_t stream) {
  const float* q    = (const float*)d_in[0];
  const float* k    = (const float*)d_in[1];
  const float* v    = (const float*)d_in[2];
  const int*   mask = (const int*)d_in[3];

  float* out   = (float*)d_out;                          // [4,16,2048,64]
  float* score = out + (size_t)4 * 16 * 2048 * 64;       // [4,16,2048,2048]

  // 2048 rows / (16 rows/wave * 8 waves) = 16 blocks in x; y = heads, z = batch.
  dim3 grid(16, 16, 4);
  sdpa_f32_wmma_kernel<<<grid, 256, 0, stream>>>(q, k, v, mask, out, score);
}